// MF_FusedForwardOps_39238821216548
// MI455X (gfx1250) — compile-verified
//
#include <hip/hip_runtime.h>
#include <cstdint>
#include <cstddef>

// MatchAttention fused forward for MI455X (gfx1250, wave32).
// One wave per (b, n, head) unit; lane == channel (Ch = 32).
// K rows async-gathered into LDS (ASYNCcnt path), V rows async-prefetched
// during similarity compute; wave32 butterfly reduction for the L1 distance.

#define WAVES_PER_BLOCK 2
#define KSTAGE_MAX 64      // window positions staged in LDS (reference K = 49)
#define K_SIM_MAX 128      // sims parked in LDS
#define CH 32              // channels per head == wave32 lanes (C=256, h=8)

constexpr int NHEADS = 8;  // fixed by reference setup_inputs (h = 8)

typedef __attribute__((address_space(1))) int as1_int;
typedef __attribute__((address_space(3))) int as3_int;

__device__ __forceinline__ void async_gather_b32(const float* gptr, float* lptr) {
#if __has_builtin(__builtin_amdgcn_global_load_async_to_lds_b32)
    __builtin_amdgcn_global_load_async_to_lds_b32((as1_int*)gptr, (as3_int*)lptr, 0, 0);
#else
    unsigned lds_addr = (unsigned)(uintptr_t)lptr;               // low 32 bits = LDS byte offset
    unsigned long long gaddr = (unsigned long long)(uintptr_t)gptr;
    asm volatile("global_load_async_to_lds_b32 %0, %1, off"
                 :
                 : "v"(lds_addr), "v"(gaddr)
                 : "memory");
#endif
}

__device__ __forceinline__ void wait_async_zero() {
#if __has_builtin(__builtin_amdgcn_s_wait_asynccnt)
    __builtin_amdgcn_s_wait_asynccnt(0);
#else
    asm volatile("s_wait_asynccnt 0" ::: "memory");
#endif
}

__global__ __launch_bounds__(WAVES_PER_BLOCK * 32)
void match_attn_fused(const float* __restrict__ moff,
                      const float* __restrict__ qg,
                      const float* __restrict__ kg,
                      const float* __restrict__ vg,
                      const int* __restrict__ pH, const int* __restrict__ pW,
                      const int* __restrict__ prh, const int* __restrict__ prw,
                      float* __restrict__ outg, float* __restrict__ attng,
                      int U)
{
    __shared__ float kbuf[WAVES_PER_BLOCK][KSTAGE_MAX][CH];
    __shared__ float vbuf[WAVES_PER_BLOCK][KSTAGE_MAX][CH];
    __shared__ float simbuf[WAVES_PER_BLOCK][K_SIM_MAX];

    const int lane = (int)(threadIdx.x & 31u);
    const int wid  = (int)(threadIdx.x >> 5u);
    const int u    = (int)blockIdx.x * WAVES_PER_BLOCK + wid;
    if (u >= U) return;

    const int H = *pH, W = *pW, rh = *prh, rw = *prw;
    const int N = H * W;
    const int K = (2 * rh + 1) * (2 * rw + 1);
    const int C = NHEADS * CH;                // 256 floats per pixel row

    const int rowq = u / NHEADS;              // b*N + n  (NHEADS pow2 -> shift)
    const int n    = rowq % N;
    const int y    = n / W;
    const int x    = n - y * W;

    // jnp.round == round-half-to-even == __float2int_rn
    const int cy = y + __float2int_rn(moff[2 * u + 0]);
    const int cx = x + __float2int_rn(moff[2 * u + 1]);

    const int eq = u * CH;                    // flat element offset of this unit's row (< 2^31)
    const float qv = qg[eq + lane];

    const int kst = (K <= KSTAGE_MAX) ? K : KSTAGE_MAX;

    // ---- async-gather K rows into LDS (one 128B coalesced row per instruction) ----
    {
        int p = 0;
        for (int dy = -rh; dy <= rh; ++dy) {
            const int py = min(max(cy + dy, 0), H - 1);
            for (int dx = -rw; dx <= rw; ++dx, ++p) {
                if (p < kst) {
                    const int px = min(max(cx + dx, 0), W - 1);
                    const int d  = (py * W + px - n) * C;
                    async_gather_b32(kg + (eq + d + lane), &kbuf[wid][p][lane]);
                }
            }
        }
    }
    wait_async_zero();

    // ---- prefetch V rows asynchronously; they complete while we compute sims ----
    {
        int p = 0;
        for (int dy = -rh; dy <= rh; ++dy) {
            const int py = min(max(cy + dy, 0), H - 1);
            for (int dx = -rw; dx <= rw; ++dx, ++p) {
                if (p < kst) {
                    const int px = min(max(cx + dx, 0), W - 1);
                    const int d  = (py * W + px - n) * C;
                    async_gather_b32(vg + (eq + d + lane), &vbuf[wid][p][lane]);
                }
            }
        }
    }

    // ---- sim_p = -sum_c |q_c - k_{p,c}|  (wave32 butterfly over lanes) ----
    float m = -3.0e38f;
    {
        int p = 0;
        for (int dy = -rh; dy <= rh; ++dy) {
            const int py = min(max(cy + dy, 0), H - 1);
            for (int dx = -rw; dx <= rw; ++dx, ++p) {
                float kvv;
                if (p < kst) {
                    kvv = kbuf[wid][p][lane];
                } else {
                    const int px = min(max(cx + dx, 0), W - 1);
                    const int d  = (py * W + px - n) * C;
                    kvv = kg[eq + d + lane];
                }
                float dsum = fabsf(qv - kvv);
                dsum += __shfl_xor(dsum, 16);
                dsum += __shfl_xor(dsum, 8);
                dsum += __shfl_xor(dsum, 4);
                dsum += __shfl_xor(dsum, 2);
                dsum += __shfl_xor(dsum, 1);
                const float sim = -dsum;                 // wave-uniform
                if (lane == 0) simbuf[wid][p] = sim;
                m = fmaxf(m, sim);
            }
        }
    }

    wait_async_zero();   // V rows now resident in LDS

    // ---- softmax + attn @ V ----
    float acc = 0.0f, ssum = 0.0f;
    {
        int p = 0;
        for (int dy = -rh; dy <= rh; ++dy) {
            const int py = min(max(cy + dy, 0), H - 1);
            for (int dx = -rw; dx <= rw; ++dx, ++p) {
                const float w = __expf(simbuf[wid][p] - m);  // wave-uniform
                float vvv;
                if (p < kst) {
                    vvv = vbuf[wid][p][lane];
                } else {
                    const int px = min(max(cx + dx, 0), W - 1);
                    const int d  = (py * W + px - n) * C;
                    vvv = vg[eq + d + lane];
                }
                ssum += w;
                acc = fmaf(w, vvv, acc);
            }
        }
    }
    const float inv = 1.0f / ssum;
    outg[eq + lane] = acc * inv;

    // attention weights: K contiguous floats per unit, lanes strided
    for (int p = lane; p < K; p += 32) {
        attng[(size_t)u * K + p] = __expf(simbuf[wid][p] - m) * inv;
    }
}

extern "C" void kernel_launch(void* const* d_in, const int* in_sizes, int n_in,
                              void* d_out, int out_size, void* d_ws, size_t ws_size,
                              hipStream_t stream) {
    const float* moff = (const float*)d_in[0];   // [B,N,h,2]
    const float* q    = (const float*)d_in[1];   // [B,N,C]
    const float* k    = (const float*)d_in[2];
    const float* v    = (const float*)d_in[3];
    const int* pH  = (const int*)d_in[4];
    const int* pW  = (const int*)d_in[5];
    const int* prh = (const int*)d_in[6];
    const int* prw = (const int*)d_in[7];

    const int U = in_sizes[0] / 2;   // B*N*h query-head units (65536)

    float* out  = (float*)d_out;                       // [B,N,C]
    float* attn = out + (size_t)in_sizes[1];           // [B,N,h,K] after B*N*C floats

    const int blocks = (U + WAVES_PER_BLOCK - 1) / WAVES_PER_BLOCK;
    hipLaunchKernelGGL(match_attn_fused, dim3(blocks), dim3(WAVES_PER_BLOCK * 32), 0, stream,
                       moff, q, k, v, pH, pW, prh, prw, out, attn, U);
}